// NaiveBayes_19275813225052
// MI455X (gfx1250) — compile-verified
//
#include <hip/hip_runtime.h>
#include <hip/hip_bf16.h>

// NaiveBayes classifier, MI455X (gfx1250, wave32).
// Memory-bound kernel (~100 MB traffic -> ~4.3 us at 23.3 TB/s). The per-class
// Gaussian log-likelihood is expressed as a [rows x 64] x [64 x 8] f32 GEMM and
// executed on the matrix pipe via chained V_WMMA_F32_16X16X4_F32 (f32 in/out,
// so numerics match the f32 reference). Data is streamed through LDS with
// coalesced b128 loads/stores.

#define NB_F        20          // features
#define NB_C        8           // classes
#define NB_XCOLS    40          // feats + mask per row
#define NB_ROWS_PB  128         // rows per block (8 waves x 16 rows)
#define NB_XPITCH   44          // LDS pitch for x tile (16B aligned, low conflicts)
#define NB_PPITCH   9           // LDS pitch for probs staging (conflict-free)

typedef __attribute__((ext_vector_type(2))) float v2f;
typedef __attribute__((ext_vector_type(4))) float v4f;
typedef __attribute__((ext_vector_type(8))) float v8f;

// B-matrix (64 x 8) coefficient of the log-space classifier GEMM:
//   k in [0,20)  : weight of m_f * x_f^2   -> -1/(2 sigma^2) on Gaussian cols
//   k in [20,40) : weight of m_f * x_f     ->  mu/sigma^2
//   k in [40,60) : weight of m_f           ->  ln(inv_norm) - mu/(2 sigma^2)  (mu^2 == mu)
//   k in [60,64) : zero padding
__device__ __forceinline__ float nb_coef(int k, int n) {
  if (n >= NB_C) return 0.0f;
  int f, which;
  if (k < 20)      { f = k;      which = 0; }
  else if (k < 40) { f = k - 20; which = 1; }
  else if (k < 60) { f = k - 40; which = 2; }
  else return 0.0f;
  if (f < n || f > n + 2) return 0.0f;                 // gaussian cols are n..n+2
  const float inv_s2      = 1.0f / (0.3f * 0.3f);      // 1/sigma^2
  const float inv_2s2     = 0.5f * inv_s2;             // 1/(2 sigma^2)
  const float ln_inv_norm = 0.28503452f;               // ln(1/(sigma*sqrt(2*pi)))
  float mu = (float)((n >> (f - n)) & 1);              // mus[n] = [n&1,(n>>1)&1,(n>>2)&1]
  if (which == 0) return -inv_2s2;
  if (which == 1) return mu * inv_s2;
  return ln_inv_norm - mu * inv_2s2;
}

__global__ __launch_bounds__(256) void nb_wmma_kernel(const float* __restrict__ x,
                                                      float* __restrict__ out) {
  __shared__ float    xs[NB_ROWS_PB * NB_XPITCH];   // 22528 B
  __shared__ float    ps[NB_ROWS_PB * NB_PPITCH];   //  4608 B
  __shared__ unsigned zp[NB_ROWS_PB * 2];           //  1024 B  (zero-mask halves)

  const int tid  = threadIdx.x;
  const int lane = tid & 31;
  const int wave = tid >> 5;
  const long long rowBase = (long long)blockIdx.x * NB_ROWS_PB;

  // ---- Phase 1: coalesced b128 stream of the 128x40 f32 tile into LDS ------
  const v4f* __restrict__ xin = (const v4f*)(x + rowBase * NB_XCOLS);
  #pragma unroll
  for (int i = 0; i < 5; ++i) {
    int idx4 = tid + i * 256;            // 0..1279 float4s
    int j    = idx4 * 4;                 // float index; 40 % 4 == 0 -> stays in-row
    int r    = j / NB_XCOLS;
    int c    = j - r * NB_XCOLS;
    *(v4f*)(&xs[r * NB_XPITCH + c]) = xin[idx4];
  }
  __syncthreads();

  // ---- Phase 2: per-wave 16-row tile ---------------------------------------
  // WMMA 32-bit fragment striping (wave32): lane L -> row m = L&15,
  // K within each 4-step = {2*hi, 2*hi+1} where hi = L>=16 (same for A and B).
  const int m   = lane & 15;
  const int hi  = lane >> 4;
  const int n   = m;                     // class column for B / C / D fragments
  const int wr0 = wave * 16;             // this wave's first local row

  // Per-lane slice of its row: f = 4j + 2*hi + {0,1}, j = 0..4
  float fe[10], ma[10];
  const float* rowp = &xs[(wr0 + m) * NB_XPITCH];
  #pragma unroll
  for (int j = 0; j < 5; ++j) {
    int f0 = 4 * j + 2 * hi;
    fe[2*j]   = rowp[f0];
    fe[2*j+1] = rowp[f0 + 1];
    ma[2*j]   = rowp[NB_F + f0];
    ma[2*j+1] = rowp[NB_F + f0 + 1];
  }

  // Zero-mask: bit f set iff base_m[f] == 0 (mask>0 and x outside [0,1)).
  unsigned zbits = 0u;
  #pragma unroll
  for (int j = 0; j < 10; ++j) {
    int  f  = 4 * (j >> 1) + 2 * hi + (j & 1);
    bool zb = (ma[j] > 0.0f) && !((fe[j] >= 0.0f) && (fe[j] < 1.0f));
    zbits |= (zb ? 1u : 0u) << f;
  }
  zp[(wr0 + m) * 2 + hi] = zbits;        // lane pair (L, L+16) covers all 20 bits

  // Constant B-matrix fragments for this lane (16 K-steps x 2 values).
  float bc[32];
  #pragma unroll
  for (int kk = 0; kk < 16; ++kk) {
    int k0 = 4 * kk + 2 * hi;
    bc[2*kk]   = nb_coef(k0,     n);
    bc[2*kk+1] = nb_coef(k0 + 1, n);
  }

  // 16 chained f32 WMMAs: t[16x16] = phi[16x64] * B[64x16]  (cols >= 8 are 0).
  v8f acc = {};
  #pragma unroll
  for (int kk = 0; kk < 16; ++kk) {
    v2f a, b;
    if (kk < 5)       { int q = kk;      a[0] = ma[2*q]   * fe[2*q]   * fe[2*q];
                                         a[1] = ma[2*q+1] * fe[2*q+1] * fe[2*q+1]; }
    else if (kk < 10) { int q = kk - 5;  a[0] = ma[2*q]   * fe[2*q];
                                         a[1] = ma[2*q+1] * fe[2*q+1]; }
    else if (kk < 15) { int q = kk - 10; a[0] = ma[2*q];
                                         a[1] = ma[2*q+1]; }
    else              { a[0] = 0.0f;     a[1] = 0.0f; }
    b[0] = bc[2*kk];
    b[1] = bc[2*kk+1];
    acc = __builtin_amdgcn_wmma_f32_16x16x4_f32(
        /*neg_a=*/false, a, /*neg_b=*/false, b,
        /*c_mod=*/(short)0, acc, /*reuse_a=*/false, /*reuse_b=*/false);
  }

  // ---- Phase 3: exp, ng mask, 1/C; stage probs to LDS ----------------------
  // C/D layout: VGPR v -> local row (v + 8*hi), column n = lane&15.
  if (n < NB_C) {
    #pragma unroll
    for (int v = 0; v < 8; ++v) {
      int      rl = wr0 + v + 8 * hi;
      unsigned z  = zp[rl * 2] | zp[rl * 2 + 1];
      bool     ng = ((z & ~(7u << n)) == 0u);      // all non-gaussian base_m == 1
      float    e  = __expf(acc[v]);                // g_term (== 1 when fully unmasked)
      ps[rl * NB_PPITCH + n] = ng ? e * 0.125f : 0.0f;   // * 1/C
    }
  }
  // Intra-wave LDS producer/consumer: DS ops are in-order per wave, no barrier.

  // ---- Phase 4: normalize + coalesced b128 store ---------------------------
  {
    const int rl   = wr0 + (lane >> 1);            // 2 lanes per row
    const int half = lane & 1;
    const float* pr = &ps[rl * NB_PPITCH];
    float p0 = pr[0], p1 = pr[1], p2 = pr[2], p3 = pr[3];
    float p4 = pr[4], p5 = pr[5], p6 = pr[6], p7 = pr[7];
    float s = ((p0 + p1) + (p2 + p3)) + ((p4 + p5) + (p6 + p7));
    s = fmaxf(s, 1e-10f);                          // jnp.clip(sum, 1e-10)
    float inv = 1.0f / s;
    v4f o;
    o[0] = (half ? p4 : p0) * inv;
    o[1] = (half ? p5 : p1) * inv;
    o[2] = (half ? p6 : p2) * inv;
    o[3] = (half ? p7 : p3) * inv;
    *(v4f*)(out + (rowBase + rl) * NB_C + 4 * half) = o;
  }
}

extern "C" void kernel_launch(void* const* d_in, const int* in_sizes, int n_in,
                              void* d_out, int out_size, void* d_ws, size_t ws_size,
                              hipStream_t stream) {
  (void)n_in; (void)out_size; (void)d_ws; (void)ws_size;
  const float* x   = (const float*)d_in[0];
  float*       out = (float*)d_out;
  const int nrows  = in_sizes[0] / NB_XCOLS;       // 524288
  const int blocks = nrows / NB_ROWS_PB;           // 4096 (exact: 524288 % 128 == 0)
  nb_wmma_kernel<<<blocks, 256, 0, stream>>>(x, out);
}